// EmilySAGE_angle_87703232184758
// MI455X (gfx1250) — compile-verified
//
#include <hip/hip_runtime.h>

#define N_NODES 100000
#define N_EDGES 600000
#define D 128
#define LDS_STRIDE 132   // 128 + 4 pad -> conflict-free ds_load_b64 fragments

typedef __attribute__((ext_vector_type(2))) float v2f;
typedef __attribute__((ext_vector_type(8))) float v8f;

// Types for the async global->LDS DMA builtin (signature from hipcc
// diagnostic: param 0 = 'int __vector(4) __device__ *', i.e. AS(1) int4*).
typedef int v4i_t __attribute__((vector_size(16)));
typedef __attribute__((address_space(1))) v4i_t gbl_v4i;
typedef __attribute__((address_space(3))) v4i_t lds_v4i;

#if defined(__HIP_DEVICE_COMPILE__) && \
    __has_builtin(__builtin_amdgcn_global_load_async_to_lds_b128) && \
    __has_builtin(__builtin_amdgcn_s_wait_asynccnt)
#define SAGE_HAVE_ASYNC_LDS 1
#else
#define SAGE_HAVE_ASYNC_LDS 0
#endif

// ---------------------------------------------------------------------------
// Kernel 0: zero the workspace (agg[N*D] ++ deg[N]); harness poisons ws.
// ---------------------------------------------------------------------------
__global__ void sage_zero_kernel(float4* __restrict__ ws, int n4) {
    int i = blockIdx.x * blockDim.x + threadIdx.x;
    int stride = gridDim.x * blockDim.x;
    float4 z = make_float4(0.f, 0.f, 0.f, 0.f);
    for (; i < n4; i += stride) ws[i] = z;
}

// ---------------------------------------------------------------------------
// Kernel 1: edge scatter. One wave (32 lanes) per edge; each lane moves a
// float4 (16B) of the 512B source row -> global_atomic_add_f32 into agg[dst].
// Lane 0 bumps deg[dst]. Feature table (51MB) + agg (51MB) are L2-resident.
// ---------------------------------------------------------------------------
__global__ void __launch_bounds__(256)
sage_scatter_kernel(const float* __restrict__ feat,
                    const long long* __restrict__ src_idx,
                    const long long* __restrict__ dst_idx,
                    float* __restrict__ agg,
                    float* __restrict__ deg) {
    int t = blockIdx.x * blockDim.x + threadIdx.x;
    int e = t >> 5;
    int lane = t & 31;
    if (e >= N_EDGES) return;
    long long s = src_idx[e];
    long long d = dst_idx[e];
    float4 v = ((const float4*)(feat + s * (long long)D))[lane];
    float* ad = agg + d * (long long)D + lane * 4;
    atomicAdd(ad + 0, v.x);
    atomicAdd(ad + 1, v.y);
    atomicAdd(ad + 2, v.z);
    atomicAdd(ad + 3, v.w);
    if (lane == 0) atomicAdd(deg + d, 1.0f);
}

// 16B global -> LDS stage: async DMA (ASYNCcnt) when available, else through
// VGPRs. `l` must be a pointer into __shared__ memory.
__device__ __forceinline__ void stage16(const float* g, float* l) {
#if SAGE_HAVE_ASYNC_LDS
    __builtin_amdgcn_global_load_async_to_lds_b128(
        (gbl_v4i*)g, (lds_v4i*)l, 0, 0);
#else
    *(float4*)l = *(const float4*)g;
#endif
}

// ---------------------------------------------------------------------------
// Kernel 2: fused  out = (agg * 1/max(deg,1)) @ W_l^T + b_l + feat @ W_r^T
// using V_WMMA_F32_16X16X4_F32. One wave per 16x16 output tile; 8 waves/block
// cover a 16-row x 128-col strip. The 16 shared A-rows (agg + feat) are
// staged into LDS once per block (async global->LDS when available), so the
// 8 waves feed their A fragments from ds_load_b64 instead of 8x-redundant
// global reads. Mean-normalization folded into the A load, bias into store.
//
// Fragment layouts (ISA 7.12.2, wave32):
//   A (16x4 f32): lane L -> row m = L&15; VGPR0/1 = K = k+2*(L>>4) .. +1
//   B (4x16 f32): lane L -> col n = L&15; VGPR0/1 = K = k+2*(L>>4) .. +1
//   C/D (16x16):  VGPR r -> m = r + 8*(L>>4), n = L&15
// ---------------------------------------------------------------------------
__global__ void __launch_bounds__(256)
sage_wmma_gemm_kernel(const float* __restrict__ agg,
                      const float* __restrict__ deg,
                      const float* __restrict__ feat,
                      const float* __restrict__ Wl,
                      const float* __restrict__ bl,
                      const float* __restrict__ Wr,
                      float* __restrict__ out) {
    __shared__ float aT[16 * LDS_STRIDE];
    __shared__ float fT[16 * LDS_STRIDE];

    const int lane  = threadIdx.x & 31;
    const int wave  = threadIdx.x >> 5;   // column tile 0..7
    const int mHalf = lane >> 4;          // 0: K+0..1, 1: K+2..3
    const int ml    = lane & 15;
    const int rowBase = blockIdx.x * 16;

    // ---- stage 16 rows x 512B of agg and feat into LDS (once per block) ----
    // 512 16B-chunks per matrix, 256 threads -> 2 chunks each.
    for (int ch = threadIdx.x; ch < 512; ch += 256) {
        const int r = ch >> 5;           // row 0..15
        const int c = ch & 31;           // 16B chunk 0..31 within row
        const size_t goff = (size_t)(rowBase + r) * D + c * 4;
        const int    loff = r * LDS_STRIDE + c * 4;
        stage16(agg + goff, &aT[loff]);
        stage16(feat + goff, &fT[loff]);
    }
#if SAGE_HAVE_ASYNC_LDS
    __builtin_amdgcn_s_wait_asynccnt(0);
#endif
    __syncthreads();

    const int row = rowBase + ml;             // A-matrix row for this lane
    const int col = wave * 16 + ml;           // B-matrix column / output col

    const float scale = 1.0f / fmaxf(deg[row], 1.0f);

    const float* wlRow = Wl + (size_t)col * D;   // B[k][n] = Wl[n*D + k]
    const float* wrRow = Wr + (size_t)col * D;

    v8f c0 = {};   // agg @ Wl^T accumulator
    v8f c1 = {};   // feat @ Wr^T accumulator

#pragma unroll 4
    for (int k = 0; k < D; k += 4) {
        const int kk = k + 2 * mHalf;
        v2f aA = *(const v2f*)&aT[ml * LDS_STRIDE + kk];   // ds_load_b64
        aA *= scale;                                       // fused mean
        v2f aF = *(const v2f*)&fT[ml * LDS_STRIDE + kk];
        v2f bL = *(const v2f*)(wlRow + kk);
        v2f bR = *(const v2f*)(wrRow + kk);
        c0 = __builtin_amdgcn_wmma_f32_16x16x4_f32(false, aA, false, bL,
                                                   (short)0, c0, false, false);
        c1 = __builtin_amdgcn_wmma_f32_16x16x4_f32(false, aF, false, bR,
                                                   (short)0, c1, false, false);
    }

    const float bias = bl[col];
    float* outBase = out + (size_t)rowBase * D + wave * 16;
#pragma unroll
    for (int r = 0; r < 8; ++r) {
        const int m = r + 8 * mHalf;
        outBase[(size_t)m * D + ml] = c0[r] + c1[r] + bias;
    }
}

// ---------------------------------------------------------------------------
// Launch. Inputs (setup_inputs order): feature f32[N,D], edge_index i64[2,E],
// W_l f32[D,D], b_l f32[D], W_r f32[D,D]. Output f32[N,D].
// Workspace: agg (N*D f32) ++ deg (N f32) = ~51.6 MB.
// ---------------------------------------------------------------------------
extern "C" void kernel_launch(void* const* d_in, const int* in_sizes, int n_in,
                              void* d_out, int out_size, void* d_ws, size_t ws_size,
                              hipStream_t stream) {
    const float*     feat = (const float*)d_in[0];
    const long long* eidx = (const long long*)d_in[1];
    const float*     Wl   = (const float*)d_in[2];
    const float*     bl   = (const float*)d_in[3];
    const float*     Wr   = (const float*)d_in[4];
    float*           out  = (float*)d_out;

    float* agg = (float*)d_ws;
    float* deg = agg + (size_t)N_NODES * D;

    const long long* src_idx = eidx;            // edge_index[0]
    const long long* dst_idx = eidx + N_EDGES;  // edge_index[1]

    // 1) zero agg + deg  (N*(D+1) floats, divisible by 4)
    int n4 = (N_NODES * (D + 1)) / 4;
    sage_zero_kernel<<<4096, 256, 0, stream>>>((float4*)d_ws, n4);

    // 2) scatter-add: 32 lanes per edge -> 8 edges per 256-thread block
    int scatter_blocks = (N_EDGES + 7) / 8;
    sage_scatter_kernel<<<scatter_blocks, 256, 0, stream>>>(
        feat, src_idx, dst_idx, agg, deg);

    // 3) fused mean + dual GEMM + bias via WMMA: 6250 blocks x 8 waves
    sage_wmma_gemm_kernel<<<N_NODES / 16, 256, 0, stream>>>(
        agg, deg, feat, Wl, bl, Wr, out);
}